// SpatialTransformer_77833397338118
// MI455X (gfx1250) — compile-verified
//
#include <hip/hip_runtime.h>

// 3D trilinear warp (spatial transformer), vol [2,2,128,128,128] f32,
// df [2,3,128,128,128] f32 -> out [2,2,128,128,128] f32.
//
// Memory-bound gather kernel. Volume (33.5 MB) is L2-resident (192 MB L2);
// df/out streamed with non-temporal policy. Gathers: one global_load_b64 per
// (z,y) corner row (both x-corners), SADDR-form addressing (scalar base +
// 32-bit VGPR byte offset) since the batch index is block-uniform.

typedef float v2f __attribute__((ext_vector_type(2)));
typedef float v4f __attribute__((ext_vector_type(4)));

#define NSP (1 << 21)   // 128^3 spatial voxels per batch
#define HW  (1 << 14)   // 128*128

// Unaligned (4-byte) 8-byte vector load -> global_load_b64.
__device__ __forceinline__ v2f load2u(const float* p) {
    v2f r;
    __builtin_memcpy(&r, p, sizeof(r));
    return r;
}

__global__ __launch_bounds__(256) void warp3d_trilinear_kernel(
    const float* __restrict__ vol,   // [2,2,128,128,128]
    const float* __restrict__ df,    // [2,3,128,128,128]
    float* __restrict__ out)         // [2,2,128,128,128]
{
    // 4096 blocks; 2048 blocks per batch; 1024 voxels per block (4/thread).
    const int blk = blockIdx.x;
    const int b   = blk >> 11;                                // uniform -> SGPR
    const int s   = ((blk & 2047) << 10) + ((int)threadIdx.x << 2);
    const int x0  = s & 127;
    const int y   = (s >> 7) & 127;
    const int z   = s >> 14;

    // Scalar base pointers (b is uniform): SADDR-form vmem everywhere.
    const float* __restrict__ df0 = df + ((size_t)(b * 3) << 21);
    const float* __restrict__ df1 = df0 + NSP;
    const float* __restrict__ df2 = df1 + NSP;
    const float* __restrict__ v0  = vol + ((size_t)(b * 2) << 21);
    const float* __restrict__ v1  = v0 + NSP;
    float* __restrict__ out0      = out + ((size_t)(b * 2) << 21);
    float* __restrict__ out1      = out0 + NSP;

    // Streaming reads of the deformation field: non-temporal (keep L2 for vol).
    const v4f dz4 = __builtin_nontemporal_load((const v4f*)(df0 + s));
    const v4f dy4 = __builtin_nontemporal_load((const v4f*)(df1 + s));
    const v4f dx4 = __builtin_nontemporal_load((const v4f*)(df2 + s));

    const float zf = (float)z;
    const float yf = (float)y;

    v4f o0, o1;

#pragma unroll
    for (int j = 0; j < 4; ++j) {
        // ---- per-dimension sample locations & weights ----
        // NOTE: reference's clip(l1-l, 0, 1) is a provable no-op
        // (l in [0,127], l1 = min(floor(l)+1,127) => l1-l in [0,1]),
        // so weights are computed without the redundant clamp.
        // z
        float lz  = fminf(fmaxf(zf + dz4[j], 0.0f), 127.0f);
        float fz  = floorf(lz);
        float lz1 = fminf(fz + 1.0f, 127.0f);
        float wzA = lz1 - lz;            // weight at floor corner
        float wzB = 1.0f - wzA;          // weight at +1 corner
        int   iz0 = (int)fz, iz1 = (int)lz1;
        // y
        float ly  = fminf(fmaxf(yf + dy4[j], 0.0f), 127.0f);
        float fy  = floorf(ly);
        float ly1 = fminf(fy + 1.0f, 127.0f);
        float wyA = ly1 - ly;
        float wyB = 1.0f - wyA;
        int   iy0 = (int)fy, iy1 = (int)ly1;
        // x
        float lx  = fminf(fmaxf((float)(x0 + j) + dx4[j], 0.0f), 127.0f);
        float fx  = floorf(lx);
        float lx1 = fminf(fx + 1.0f, 127.0f);
        float wxA = lx1 - lx;
        float wxB = 1.0f - wxA;
        int   ix0 = (int)fx;

        // Paired x-load base: both corners live at [m, m+1] (elements).
        // If ix0==127 then wxA==0 exactly, so the low element (x=126)
        // contributes 0; the high element is v[127]==v[ix1]. No OOB.
        const int m = min(ix0, 126);

        // ---- 4 corner-row offsets in ELEMENTS (shared across channels) ----
        const int oz0 = iz0 << 14, oz1 = iz1 << 14;
        const int oy0 = iy0 << 7,  oy1 = iy1 << 7;
        const int r00 = oz0 + oy0 + m;
        const int r01 = oz0 + oy1 + m;
        const int r10 = oz1 + oy0 + m;
        const int r11 = oz1 + oy1 + m;

        // ---- 8 trilinear weights ----
        const float w00 = wzA * wyA, w01 = wzA * wyB;
        const float w10 = wzB * wyA, w11 = wzB * wyB;
        const float w000 = w00 * wxA, w001 = w00 * wxB;
        const float w010 = w01 * wxA, w011 = w01 * wxB;
        const float w100 = w10 * wxA, w101 = w10 * wxB;
        const float w110 = w11 * wxA, w111 = w11 * wxB;

        // ---- paired gathers: 4 b64 loads per channel, SADDR-form ----
        const v2f p00 = load2u(v0 + r00);
        const v2f p01 = load2u(v0 + r01);
        const v2f p10 = load2u(v0 + r10);
        const v2f p11 = load2u(v0 + r11);
        const v2f q00 = load2u(v1 + r00);
        const v2f q01 = load2u(v1 + r01);
        const v2f q10 = load2u(v1 + r10);
        const v2f q11 = load2u(v1 + r11);

        // ---- accumulate, reference corner order (z,y,x bits) ----
        float a0 = p00.x * w000;
        a0 = fmaf(p00.y, w001, a0);
        a0 = fmaf(p01.x, w010, a0);
        a0 = fmaf(p01.y, w011, a0);
        a0 = fmaf(p10.x, w100, a0);
        a0 = fmaf(p10.y, w101, a0);
        a0 = fmaf(p11.x, w110, a0);
        a0 = fmaf(p11.y, w111, a0);

        float a1 = q00.x * w000;
        a1 = fmaf(q00.y, w001, a1);
        a1 = fmaf(q01.x, w010, a1);
        a1 = fmaf(q01.y, w011, a1);
        a1 = fmaf(q10.x, w100, a1);
        a1 = fmaf(q10.y, w101, a1);
        a1 = fmaf(q11.x, w110, a1);
        a1 = fmaf(q11.y, w111, a1);

        o0[j] = a0;
        o1[j] = a1;
    }

    // Streaming writes: non-temporal 128-bit stores, one per channel.
    __builtin_nontemporal_store(o0, (v4f*)(out0 + s));
    __builtin_nontemporal_store(o1, (v4f*)(out1 + s));
}

extern "C" void kernel_launch(void* const* d_in, const int* in_sizes, int n_in,
                              void* d_out, int out_size, void* d_ws, size_t ws_size,
                              hipStream_t stream) {
    (void)in_sizes; (void)n_in; (void)out_size; (void)d_ws; (void)ws_size;
    const float* vol = (const float*)d_in[0];   // 2*2*128^3 f32
    const float* df  = (const float*)d_in[1];   // 2*3*128^3 f32
    float*       out = (float*)d_out;           // 2*2*128^3 f32

    // 2 * 128^3 voxels total, 4 voxels per thread -> 2^20 threads.
    const int threads = 256;                     // 8 wave32 per block
    const int blocks  = (2 * NSP / 4) / threads; // 4096
    warp3d_trilinear_kernel<<<blocks, threads, 0, stream>>>(vol, df, out);
}